// AMPGCN_60112362275099
// MI455X (gfx1250) — compile-verified
//
#include <hip/hip_runtime.h>
#include <hip/hip_bf16.h>

#define NN    20000
#define SS    20
#define DD    768
#define VOC   1433
#define NC    7
#define SD    (SS * DD)     // 15360
#define VP    1440          // vocab padded to multiple of 16
#define EPSF  1e-5f
#define WSTR  772           // padded LDS row stride for W (bank-conflict free)

typedef __attribute__((ext_vector_type(2))) float v2f;
typedef __attribute__((ext_vector_type(8))) float v8f;

// ---------------------------------------------------------------- zero cnt
__global__ void k_zero(float* __restrict__ p, int n) {
    int i = blockIdx.x * 256 + threadIdx.x;
    if (i < n) p[i] = 0.f;
}

// ---------------------------------------------------------------- histogram
// cnt[s][v] = number of nodes whose s-th sample is vocab v (exact in fp32)
__global__ void k_hist(const int* __restrict__ idx, float* __restrict__ cnt) {
    int t = blockIdx.x * 256 + threadIdx.x;
    if (t < NN * SS) {
        int v = idx[t];
        int s = t % SS;
        atomicAdd(&cnt[s * VOC + v], 1.0f);
    }
}

// ---------------------------------------------------------------- BN stats
// Each thread owns one channel (s,d). Three sequential weighted-stat passes
// over the 1433 vocab rows produce the composed affine coefficients
// a1,c1,a2,c2,a3,c3 so that T3 = relu(a3*relu(a2*relu(a1*e+c1)+c2)+c3).
__global__ void __launch_bounds__(256)
k_stats(const float* __restrict__ emb, const float* __restrict__ cnt,
        const float* __restrict__ g1, const float* __restrict__ b1,
        const float* __restrict__ g2, const float* __restrict__ b2,
        const float* __restrict__ g3, const float* __restrict__ b3,
        float* __restrict__ coef) {
    __shared__ float sc[VOC];
    const int s = blockIdx.y;
    const int d = blockIdx.x * 256 + threadIdx.x;
    for (int v = threadIdx.x; v < VOC; v += 256) sc[v] = cnt[s * VOC + v];
    __syncthreads();

    const int ch = s * DD + d;
    const float invN = 1.0f / (float)NN;

    float m = 0.f, q = 0.f;
    for (int v = 0; v < VOC; ++v) {
        float c = sc[v];
        float e = emb[v * DD + d];
        m = fmaf(c, e, m);
        q = fmaf(c * e, e, q);
    }
    m *= invN; q *= invN;
    float a1 = g1[ch] * rsqrtf(fmaxf(q - m * m, 0.f) + EPSF);
    float c1 = fmaf(-m, a1, b1[ch]);

    m = 0.f; q = 0.f;
    for (int v = 0; v < VOC; ++v) {
        float c = sc[v];
        float t = fmaxf(fmaf(a1, emb[v * DD + d], c1), 0.f);
        m = fmaf(c, t, m);
        q = fmaf(c * t, t, q);
    }
    m *= invN; q *= invN;
    float a2 = g2[ch] * rsqrtf(fmaxf(q - m * m, 0.f) + EPSF);
    float c2 = fmaf(-m, a2, b2[ch]);

    m = 0.f; q = 0.f;
    for (int v = 0; v < VOC; ++v) {
        float c = sc[v];
        float t = fmaxf(fmaf(a1, emb[v * DD + d], c1), 0.f);
        t = fmaxf(fmaf(a2, t, c2), 0.f);
        m = fmaf(c, t, m);
        q = fmaf(c * t, t, q);
    }
    m *= invN; q *= invN;
    float a3 = g3[ch] * rsqrtf(fmaxf(q - m * m, 0.f) + EPSF);
    float c3 = fmaf(-m, a3, b3[ch]);

    coef[0 * SD + ch] = a1;
    coef[1 * SD + ch] = c1;
    coef[2 * SD + ch] = a2;
    coef[3 * SD + ch] = c2;
    coef[4 * SD + ch] = a3;
    coef[5 * SD + ch] = c3;
}

// ---------------------------------------------------------------- U GEMM (WMMA)
// U[s,v,k] = sum_d T3[s,v,d] * lin_w[k,d], via V_WMMA_F32_16X16X4_F32.
// One wave per 16-row M-tile (tiles never cross an s boundary since VP=1440).
// Per 64-wide K-chunk: transform emb tile -> LDS (stride 68, conflict-free),
// then 16 WMMA steps with B-fragments from LDS-resident lin_w.
// Branch-free everywhere except the final store: out-of-vocab pad rows are
// clamped (their U entries are never gathered), and the N>=7 B columns are
// zeroed with a select multiplier, so EXEC stays all-ones for the WMMAs.
__global__ void __launch_bounds__(256)
k_ugemm(const float* __restrict__ emb, const float* __restrict__ coef,
        const float* __restrict__ W, float* __restrict__ U) {
    __shared__ float tlds[8 * 16 * 68];   // per-wave 16x64 staging, stride 68
    __shared__ float wlds[NC * WSTR];     // lin_w, padded row stride

    // cooperative preload of lin_w into LDS (coalesced)
    for (int d = threadIdx.x; d < DD; d += 256) {
        #pragma unroll
        for (int n = 0; n < NC; ++n)
            wlds[n * WSTR + d] = W[n * DD + d];
    }
    __syncthreads();

    const int lane = threadIdx.x & 31;
    const int wave = threadIdx.x >> 5;
    const int tileId = blockIdx.x * 8 + wave;       // 0..1799
    const int s  = tileId / (VP / 16);
    const int v0 = (tileId % (VP / 16)) * 16;
    float* myT = &tlds[wave * (16 * 68)];
    const float* cf = coef + s * DD;
    const int l16  = lane & 15;
    const int half = lane >> 4;
    const float sel = (l16 < NC) ? 1.f : 0.f;            // kill B columns 7..15
    const float* myW = &wlds[(l16 < NC ? l16 : 0) * WSTR];

    v8f acc = {0.f, 0.f, 0.f, 0.f, 0.f, 0.f, 0.f, 0.f};

    for (int cb = 0; cb < DD / 64; ++cb) {
        const int dbase = cb * 64;
        // --- fill + transform: 16 rows x 64 cols, float4 granularity -------
        #pragma unroll
        for (int i = 0; i < 8; ++i) {
            int f = i * 32 + lane;          // 0..255 float4 slots
            int r = f >> 4;                 // row 0..15
            int c = (f & 15) << 2;          // col 0..60 step 4
            int v = v0 + r;
            v = (v < VOC) ? v : (VOC - 1);  // clamp: pad rows unused downstream
            float4 e = *reinterpret_cast<const float4*>(&emb[v * DD + dbase + c]);
            const float* cp = cf + dbase + c;
            float4 A1 = *reinterpret_cast<const float4*>(cp + 0 * SD);
            float4 C1 = *reinterpret_cast<const float4*>(cp + 1 * SD);
            float4 A2 = *reinterpret_cast<const float4*>(cp + 2 * SD);
            float4 C2 = *reinterpret_cast<const float4*>(cp + 3 * SD);
            float4 A3 = *reinterpret_cast<const float4*>(cp + 4 * SD);
            float4 C3 = *reinterpret_cast<const float4*>(cp + 5 * SD);
            float4 t;
            t.x = fmaxf(fmaf(A3.x, fmaxf(fmaf(A2.x, fmaxf(fmaf(A1.x, e.x, C1.x), 0.f), C2.x), 0.f), C3.x), 0.f);
            t.y = fmaxf(fmaf(A3.y, fmaxf(fmaf(A2.y, fmaxf(fmaf(A1.y, e.y, C1.y), 0.f), C2.y), 0.f), C3.y), 0.f);
            t.z = fmaxf(fmaf(A3.z, fmaxf(fmaf(A2.z, fmaxf(fmaf(A1.z, e.z, C1.z), 0.f), C2.z), 0.f), C3.z), 0.f);
            t.w = fmaxf(fmaf(A3.w, fmaxf(fmaf(A2.w, fmaxf(fmaf(A1.w, e.w, C1.w), 0.f), C2.w), 0.f), C3.w), 0.f);
            *reinterpret_cast<float4*>(&myT[r * 68 + c]) = t;
        }
        // same-wave DS ops are in-order; compiler inserts the return waits.
        // --- 16 WMMA steps over this K-chunk ------------------------------
        #pragma unroll
        for (int kk = 0; kk < 16; ++kk) {
            int dl = kk * 4 + half * 2;
            v2f a, b;
            // A 16x4 f32 layout: lane = M, VGPR+half-select = K
            a.x = myT[l16 * 68 + dl];
            a.y = myT[l16 * 68 + dl + 1];
            // B 4x16 f32 layout: lane = N, VGPR+half-select = K; B[k][n] = W[n][d]
            int dg = dbase + dl;
            b.x = sel * myW[dg];
            b.y = sel * myW[dg + 1];
            acc = __builtin_amdgcn_wmma_f32_16x16x4_f32(
                false, a, false, b, (short)0, acc, false, false);
        }
    }

    // D layout: VGPR r -> M = r + half*8, lane -> N = l16. Store N<8 (col 7..15 == 0).
    if (l16 < 8) {
        int n  = l16;
        int mb = half * 8;
        #pragma unroll
        for (int r = 0; r < 8; ++r)
            U[(s * VP + v0 + mb + r) * 8 + n] = acc[r];
    }
}

// ---------------------------------------------------------------- gather + log_softmax
__global__ void k_logits(const int* __restrict__ idx, const float* __restrict__ U,
                         const float* __restrict__ lb, float* __restrict__ out) {
    int n = blockIdx.x * 256 + threadIdx.x;
    if (n >= NN) return;
    float acc[NC] = {0.f, 0.f, 0.f, 0.f, 0.f, 0.f, 0.f};
    for (int s = 0; s < SS; ++s) {
        int v = idx[n * SS + s];
        const float4* u = reinterpret_cast<const float4*>(&U[(s * VP + v) * 8]);
        float4 u0 = u[0];
        float4 u1 = u[1];
        acc[0] += u0.x; acc[1] += u0.y; acc[2] += u0.z; acc[3] += u0.w;
        acc[4] += u1.x; acc[5] += u1.y; acc[6] += u1.z;
    }
    float lg[NC];
    float mx = -1e30f;
    #pragma unroll
    for (int k = 0; k < NC; ++k) {
        lg[k] = acc[k] * (1.0f / (float)SS) + lb[k];
        mx = fmaxf(mx, lg[k]);
    }
    float se = 0.f;
    #pragma unroll
    for (int k = 0; k < NC; ++k) se += expf(lg[k] - mx);
    float lse = mx + logf(se);
    #pragma unroll
    for (int k = 0; k < NC; ++k) out[n * NC + k] = lg[k] - lse;
}

// ---------------------------------------------------------------- launch
extern "C" void kernel_launch(void* const* d_in, const int* in_sizes, int n_in,
                              void* d_out, int out_size, void* d_ws, size_t ws_size,
                              hipStream_t stream) {
    (void)in_sizes; (void)n_in; (void)out_size; (void)ws_size;
    const int*   idx = (const int*)  d_in[0];
    const float* emb = (const float*)d_in[1];
    const float* g1  = (const float*)d_in[2];
    const float* b1  = (const float*)d_in[3];
    const float* g2  = (const float*)d_in[4];
    const float* b2  = (const float*)d_in[5];
    const float* g3  = (const float*)d_in[6];
    const float* b3  = (const float*)d_in[7];
    const float* W   = (const float*)d_in[8];
    const float* lb  = (const float*)d_in[9];
    float* out = (float*)d_out;

    char* ws   = (char*)d_ws;
    float* cnt  = (float*)(ws);             // 28,660 floats (rounded region 114,688 B)
    float* coef = (float*)(ws + 114688);    // 6 * 15,360 floats = 368,640 B
    float* U    = (float*)(ws + 483328);    // 20*1440*8 floats = 921,600 B

    k_zero  <<<(SS * VOC + 255) / 256, 256, 0, stream>>>(cnt, SS * VOC);
    k_hist  <<<(NN * SS + 255) / 256, 256, 0, stream>>>(idx, cnt);
    k_stats <<<dim3(DD / 256, SS), 256, 0, stream>>>(emb, cnt, g1, b1, g2, b2, g3, b3, coef);
    k_ugemm <<<(SS * (VP / 16)) / 8, 256, 0, stream>>>(emb, coef, W, U);
    k_logits<<<(NN + 255) / 256, 256, 0, stream>>>(idx, U, lb, out);
}